// Encoder_64879775973480
// MI455X (gfx1250) — compile-verified
//
#include <hip/hip_runtime.h>
#include <hip/hip_bf16.h>
#include <stdint.h>

// Encoder gather kernel for MI455X (gfx1250, wave32).
//
// out[t, 0:300]   = tok < n_pre ? vectors[tok] : 0
// out[t, 300:428] = b + (tok >= n_pre ? W[:, tok - n_pre] : 0)
//
// Memory-bound gather (~33 MB @ 23.3 TB/s ~ 1.5-2 us). One wave per token:
//  - x1: 75 x float4 coalesced row copy (global_load_b128 / global_store_b128)
//  - x2: bias staged in LDS via global_load_async_to_lds_b32 (ASYNCcnt path),
//        W-column gather (stride 50000 floats) latency-hidden with
//        global_prefetch_b8 issued before the x1 copy.

#define VEC_DIM 300
#define HIDDEN  128
#define OUT_DIM (VEC_DIM + HIDDEN)     // 428 floats = 107 float4 (16B aligned)
#define WAVES_PER_BLOCK 8
#define BLOCK_THREADS   256            // 8 wave32s

#if defined(__gfx1250__) && \
    __has_builtin(__builtin_amdgcn_global_load_async_to_lds_b32) && \
    __has_builtin(__builtin_amdgcn_s_wait_asynccnt)
#define USE_ASYNC_LDS 1
#else
#define USE_ASYNC_LDS 0
#endif

#if USE_ASYNC_LDS
// Probe-derived: first param of __builtin_amdgcn_global_load_async_to_lds_b32
// is a non-const AS1 'int*' (hipcc diagnostic round 1). Assume AS3 'int*' for
// the LDS destination.
typedef __attribute__((address_space(1))) int* gptr_i32;
typedef __attribute__((address_space(3))) int* lptr_i32;
#endif

__global__ __launch_bounds__(BLOCK_THREADS)
void encoder_gather_kernel(const int*   __restrict__ tokens,
                           const float* __restrict__ vectors,
                           const float* __restrict__ W,
                           const float* __restrict__ b,
                           float*       __restrict__ out,
                           int n_tokens, int n_pre, int in_size)
{
    __shared__ __align__(16) float s_b[HIDDEN];

    // ---- Stage bias (128 f32) into LDS, once per workgroup ----
#if USE_ASYNC_LDS
    if (threadIdx.x < HIDDEN) {
        // global -> LDS direct, tracked by ASYNCcnt (gfx1250 async path).
        // AS1 cast is bit-identical for global pointers; generic LDS pointer
        // truncates to the 32-bit LDS offset for AS3.
        gptr_i32 g = (gptr_i32)(uintptr_t)(b + threadIdx.x);
        lptr_i32 l = (lptr_i32)(uint32_t)(uintptr_t)(&s_b[threadIdx.x]);
        __builtin_amdgcn_global_load_async_to_lds_b32(g, l, 0, 0);
    }
    __builtin_amdgcn_s_wait_asynccnt(0);   // s_wait_asynccnt 0
#else
    if (threadIdx.x < HIDDEN) s_b[threadIdx.x] = b[threadIdx.x];
#endif
    __syncthreads();

    const int lane = threadIdx.x & 31;     // wave32
    const int wv   = threadIdx.x >> 5;
    const int t    = blockIdx.x * WAVES_PER_BLOCK + wv;
    if (t >= n_tokens) return;

    const int  tok    = tokens[t];          // wave-uniform
    const bool is_pre = (tok < n_pre);

    float4* __restrict__ out4 = (float4*)(out + (size_t)t * OUT_DIM);

    // ---- Column of W for OOV tokens: prefetch the 128 scattered lines early
    int col = tok - n_pre;
    col = col < 0 ? 0 : (col > in_size - 1 ? in_size - 1 : col);
    const float* __restrict__ wc = W + col;
    const int h = lane * 4;
    if (!is_pre) {
        // stride between rows is in_size (=50000) floats -> one cacheline each;
        // global_prefetch_b8 pulls them while we do the x1 stores below.
        __builtin_prefetch(wc + (size_t)(h + 0) * in_size, 0, 0);
        __builtin_prefetch(wc + (size_t)(h + 1) * in_size, 0, 0);
        __builtin_prefetch(wc + (size_t)(h + 2) * in_size, 0, 0);
        __builtin_prefetch(wc + (size_t)(h + 3) * in_size, 0, 0);
    }

    // ---- x1: 300 floats = 75 float4, coalesced b128 copy or zero-fill ----
    if (is_pre) {
        const float4* __restrict__ v4 =
            (const float4*)(vectors + (size_t)tok * VEC_DIM);  // 1200B rows, 16B aligned
        out4[lane]      = v4[lane];
        out4[lane + 32] = v4[lane + 32];
        if (lane < 75 - 64) out4[lane + 64] = v4[lane + 64];
    } else {
        const float4 z = make_float4(0.f, 0.f, 0.f, 0.f);
        out4[lane]      = z;
        out4[lane + 32] = z;
        if (lane < 75 - 64) out4[lane + 64] = z;
    }

    // ---- x2: 128 floats = 32 float4 (one per lane): bias (+ W column) ----
    const float4* sb4 = (const float4*)s_b;
    float4 r = sb4[lane];                   // ds_load_b128
    if (!is_pre) {
        r.x += wc[(size_t)(h + 0) * in_size];
        r.y += wc[(size_t)(h + 1) * in_size];
        r.z += wc[(size_t)(h + 2) * in_size];
        r.w += wc[(size_t)(h + 3) * in_size];
    }
    out4[75 + lane] = r;                    // global_store_b128
}

extern "C" void kernel_launch(void* const* d_in, const int* in_sizes, int n_in,
                              void* d_out, int out_size, void* d_ws, size_t ws_size,
                              hipStream_t stream) {
    const int*   tokens  = (const int*)  d_in[0];   // [64*200] int32
    const float* vectors = (const float*)d_in[1];   // [n_pre, 300] f32
    const float* W       = (const float*)d_in[2];   // [128, in_size] f32
    const float* b       = (const float*)d_in[3];   // [128] f32
    float*       out     = (float*)d_out;           // [n_tokens, 428] f32

    const int n_tokens = in_sizes[0];
    const int n_pre    = in_sizes[1] / VEC_DIM;
    const int in_size  = in_sizes[2] / HIDDEN;

    const int blocks = (n_tokens + WAVES_PER_BLOCK - 1) / WAVES_PER_BLOCK;
    encoder_gather_kernel<<<blocks, BLOCK_THREADS, 0, stream>>>(
        tokens, vectors, W, b, out, n_tokens, n_pre, in_size);
}